// CSM_62216896250023
// MI455X (gfx1250) — compile-verified
//
#include <hip/hip_runtime.h>
#include <hip/hip_bf16.h>
#include <stdint.h>

#define TPB    256
#define EMB    1024
#define SEQL   7
#define ITERS  8          // batch rows per block (16384 / 8 = 2048 blocks)

typedef float v4f __attribute__((ext_vector_type(4)));
typedef int   v4i __attribute__((ext_vector_type(4)));

typedef __attribute__((address_space(1))) v4i* gv4i_p;   // global B128 payload
typedef __attribute__((address_space(3))) v4i* lv4i_p;   // LDS B128 payload

// ---------------------------------------------------------------------------
// gfx1250 async global->LDS copy (ASYNCcnt-tracked DMA, bypasses VGPRs).
// ---------------------------------------------------------------------------
__device__ __forceinline__ void async_cp16(void* lds_dst, const void* gsrc) {
#if __has_builtin(__builtin_amdgcn_global_load_async_to_lds_b128)
  __builtin_amdgcn_global_load_async_to_lds_b128(
      (gv4i_p)(uintptr_t)gsrc,
      (lv4i_p)(uintptr_t)lds_dst,
      /*offset=*/0, /*cpol=*/0);
#else
  uint32_t lds_off = (uint32_t)(uintptr_t)lds_dst;
  asm volatile("global_load_async_to_lds_b128 %0, %1, off"
               :: "v"(lds_off), "v"(gsrc) : "memory");
#endif
}

// Async loads complete in order: waiting for <=7 outstanding means everything
// issued before the most recent 7 copies has landed in LDS.
__device__ __forceinline__ void wait_async_le7() {
#if __has_builtin(__builtin_amdgcn_s_wait_asynccnt)
  __builtin_amdgcn_s_wait_asynccnt(7);
#else
  asm volatile("s_wait_asynccnt 7" ::: "memory");
#endif
}

__device__ __forceinline__ void wait_async0() {
#if __has_builtin(__builtin_amdgcn_s_wait_asynccnt)
  __builtin_amdgcn_s_wait_asynccnt(0);
#else
  asm volatile("s_wait_asynccnt 0" ::: "memory");
#endif
}

// sigmoid(x) = 1 / (1 + exp2(-x * log2 e))
__device__ __forceinline__ float sig1(float x) {
  float e = __builtin_amdgcn_exp2f(x * -1.44269504088896340736f);
  return __builtin_amdgcn_rcpf(1.0f + e);
}

__device__ __forceinline__ v4f f4sig(v4f x) {
  v4f r;
  r.x = sig1(x.x); r.y = sig1(x.y); r.z = sig1(x.z); r.w = sig1(x.w);
  return r;
}

// ---------------------------------------------------------------------------
// Each block owns ITERS consecutive batch rows. 256 threads x 4 channels
// (one float4) cover the 1024-wide embedding. Per item, the 7 gathered
// embedding rows (28 KB) are DMA'd into one of two LDS buffers; the conv
// chain runs register-resident on the other buffer while the DMA engine
// fills the next one. Weights (40 VGPRs) are fetched once per block.
// ---------------------------------------------------------------------------
__global__ __launch_bounds__(TPB) void CSM_62216896250023_kernel(
    const int*   __restrict__ X,
    const float* __restrict__ emb,
    const float* __restrict__ c1, const float* __restrict__ c2,
    const float* __restrict__ c3, const float* __restrict__ c4,
    float*       __restrict__ out)
{
  __shared__ __align__(16) float lds[2][SEQL * EMB];   // 2 x 28 KB of 320 KB/WGP

  const int    tid = threadIdx.x;
  const int    c   = tid << 2;                         // first of 4 channels
  const size_t b0  = (size_t)blockIdx.x * ITERS;

  // Prologue: DMA item 0 into buffer 0.
  {
    const int* __restrict__ xr = X + b0 * SEQL;        // uniform -> s_load
#pragma unroll
    for (int t = 0; t < SEQL; ++t) {
      const size_t row = (size_t)xr[t];
      async_cp16(&lds[0][t * EMB + c], emb + row * EMB + c);
    }
  }

  // Block-invariant weights: fetched once, live in VGPRs for all ITERS items.
  const v4f w10 = *(const v4f*)(c1 + 0 * EMB + c);
  const v4f w11 = *(const v4f*)(c1 + 1 * EMB + c);
  const v4f w20 = *(const v4f*)(c2 + 0 * EMB + c);
  const v4f w21 = *(const v4f*)(c2 + 1 * EMB + c);
  const v4f w30 = *(const v4f*)(c3 + 0 * EMB + c);
  const v4f w31 = *(const v4f*)(c3 + 1 * EMB + c);
  const v4f w32 = *(const v4f*)(c3 + 2 * EMB + c);
  const v4f w40 = *(const v4f*)(c4 + 0 * EMB + c);
  const v4f w41 = *(const v4f*)(c4 + 1 * EMB + c);
  const v4f w42 = *(const v4f*)(c4 + 2 * EMB + c);

#pragma unroll
  for (int i = 0; i < ITERS; ++i) {
    // Issue next item's DMA into the other buffer (its last readers finished
    // at iteration i-1; the end-of-loop barrier ordered them before us).
    if (i + 1 < ITERS) {
      const int* __restrict__ xr = X + (b0 + i + 1) * SEQL;
#pragma unroll
      for (int t = 0; t < SEQL; ++t) {
        const size_t row = (size_t)xr[t];
        async_cp16(&lds[(i + 1) & 1][t * EMB + c], emb + row * EMB + c);
      }
      wait_async_le7();   // current buffer's 7 copies done; next 7 in flight
    } else {
      wait_async0();      // last item: drain everything
    }
    __syncthreads();      // every wave's copies for buf[i&1] have landed

    const float* __restrict__ buf = lds[i & 1];
    v4f h[SEQL];
#pragma unroll
    for (int t = 0; t < SEQL; ++t)
      h[t] = *(const v4f*)&buf[t * EMB + c];

    // conv1 (h=2), sigmoid: 7 -> 6
    v4f a[6];
#pragma unroll
    for (int p = 0; p < 6; ++p)
      a[p] = f4sig(h[p] * w10 + h[p + 1] * w11);

    // conv2 (h=2), sigmoid: 6 -> 5
    v4f s[5];
#pragma unroll
    for (int p = 0; p < 5; ++p)
      s[p] = f4sig(a[p] * w20 + a[p + 1] * w21);

    // conv3 (h=3), sigmoid: 5 -> 3
    v4f d[3];
#pragma unroll
    for (int p = 0; p < 3; ++p)
      d[p] = f4sig(s[p] * w30 + s[p + 1] * w31 + s[p + 2] * w32);

    // conv4 (h=3), sigmoid: 3 -> 1
    const v4f o = f4sig(d[0] * w40 + d[1] * w41 + d[2] * w42);

    // Output is write-once, never re-read: stream it past the caches.
    __builtin_nontemporal_store(o, (v4f*)(out + (b0 + i) * EMB + c));

    __syncthreads();      // all waves done reading buf[i&1]; DMA may reuse it
  }
}

extern "C" void kernel_launch(void* const* d_in, const int* in_sizes, int n_in,
                              void* d_out, int out_size, void* d_ws, size_t ws_size,
                              hipStream_t stream) {
  const int*   X   = (const int*)  d_in[0];
  const float* emb = (const float*)d_in[1];
  const float* c1  = (const float*)d_in[2];
  const float* c2  = (const float*)d_in[3];
  const float* c3  = (const float*)d_in[4];
  const float* c4  = (const float*)d_in[5];
  float* out = (float*)d_out;

  const int B = in_sizes[0] / SEQL;     // 16384
  const int G = B / ITERS;              // 2048 blocks
  CSM_62216896250023_kernel<<<G, TPB, 0, stream>>>(X, emb, c1, c2, c3, c4, out);
}